// HeteroSAGEAttentionConv_59854664237757
// MI455X (gfx1250) — compile-verified
//
#include <hip/hip_runtime.h>
#include <hip/hip_bf16.h>

typedef float v2f __attribute__((ext_vector_type(2)));
typedef float v8f __attribute__((ext_vector_type(8)));

// ---------------------------------------------------------------------------
// Projection GEMM: Y[n,128] = X[n,128] @ W[128,128], f32, via WMMA 16x16x4 f32.
// Block = 256 threads = 8 waves. Block handles a 16-row tile; wave w handles
// output columns [16w, 16w+16). 32 WMMA steps cover K=128.
//
// ISA 7.12.2 layouts (wave32):
//   A 16x4 (MxK): lanes 0-15 -> M=lane, {v0,v1}={K0,K1}; lanes 16-31 -> {K2,K3}
//   B 4x16 (KxN): lanes 0-15 -> N=lane, {v0,v1}={K0,K1}; lanes 16-31 -> {K2,K3}
//   C/D 16x16:    VGPR r: lanes 0-15 -> M=r, N=lane; lanes 16-31 -> M=r+8
// ---------------------------------------------------------------------------
__global__ __launch_bounds__(256) void proj_gemm_kernel(
    const float* __restrict__ X, const float* __restrict__ W,
    float* __restrict__ Y, int nrows) {
  const int lane = threadIdx.x & 31;
  const int wave = threadIdx.x >> 5;           // 0..7 -> 16-col tile
  const int row0 = blockIdx.x * 16;
  if (row0 >= nrows) return;
  const int colBase = wave * 16;
  const int m  = lane & 15;                    // row (A) / col (B) within tile
  const int kh = (lane >> 4) * 2;              // K sub-offset: 0 or 2

  const float* __restrict__ xrow = X + (size_t)(row0 + m) * 128;

  v8f c = {};
#pragma unroll
  for (int k = 0; k < 32; ++k) {
    const int kk = 4 * k + kh;
    v2f a;
    a.x = xrow[kk];
    a.y = xrow[kk + 1];
    v2f b;
    b.x = W[(size_t)kk * 128 + colBase + m];
    b.y = W[(size_t)(kk + 1) * 128 + colBase + m];
    // 8 args: (neg_a, A, neg_b, B, c_mod, C, reuse_a, reuse_b)
    c = __builtin_amdgcn_wmma_f32_16x16x4_f32(false, a, false, b, (short)0, c,
                                              false, false);
  }

  const int rbase = (lane < 16) ? 0 : 8;
#pragma unroll
  for (int r = 0; r < 8; ++r) {
    Y[(size_t)(row0 + rbase + r) * 128 + colBase + m] = c[r];
  }
}

// ---------------------------------------------------------------------------
// Edge phase: one wave per edge. Lane l owns elements 4l..4l+3 (128 = 32*4).
//   src = si[si[e]] (faithful double gather), dst = ti[e]
//   score = [s2 || t2] . a ; att = exp(leakyrelu(score, 0.2))
//   num[dst] += s2*att (f32 atomics, L2-resident), den[dst] += att
// ---------------------------------------------------------------------------
__global__ __launch_bounds__(256) void edge_kernel(
    const float* __restrict__ s, const float* __restrict__ t,
    const int* __restrict__ si, const int* __restrict__ ti,
    const float* __restrict__ a, float* __restrict__ num,
    float* __restrict__ den, int E) {
  const int lane = threadIdx.x & 31;
  const int e = (int)((blockIdx.x * blockDim.x + threadIdx.x) >> 5);
  if (e >= E) return;

  const int s1  = si[e];
  const int src = si[s1];          // s2 = s[si][si]  ->  s[si[si[e]]]
  const int dst = ti[e];

  const float4 sv = *(const float4*)(s + (size_t)src * 128 + lane * 4);
  const float4 tv = *(const float4*)(t + (size_t)dst * 128 + lane * 4);
  const float4 as = *(const float4*)(a + lane * 4);
  const float4 at = *(const float4*)(a + 128 + lane * 4);

  float p = sv.x * as.x + sv.y * as.y + sv.z * as.z + sv.w * as.w +
            tv.x * at.x + tv.y * at.y + tv.z * at.z + tv.w * at.w;
#pragma unroll
  for (int mask = 16; mask >= 1; mask >>= 1) p += __shfl_xor(p, mask, 32);

  const float att = __expf(p >= 0.0f ? p : 0.2f * p);

  float* np = num + (size_t)dst * 128 + lane * 4;
  atomicAdd(np + 0, sv.x * att);
  atomicAdd(np + 1, sv.y * att);
  atomicAdd(np + 2, sv.z * att);
  atomicAdd(np + 3, sv.w * att);
  if (lane == 0) atomicAdd(den + dst, att);
}

// ---------------------------------------------------------------------------
// Finalize: out = relu(t + num / (den + 1e-6))
// ---------------------------------------------------------------------------
__global__ __launch_bounds__(256) void finalize_kernel(
    const float* __restrict__ t, const float* __restrict__ num,
    const float* __restrict__ den, float* __restrict__ out, int total) {
  const int idx = blockIdx.x * blockDim.x + threadIdx.x;
  if (idx >= total) return;
  const int row = idx >> 7;
  const float v = t[idx] + num[idx] / (den[row] + 1e-6f);
  out[idx] = v > 0.0f ? v : 0.0f;
}

extern "C" void kernel_launch(void* const* d_in, const int* in_sizes, int n_in,
                              void* d_out, int out_size, void* d_ws,
                              size_t ws_size, hipStream_t stream) {
  const float* x_user = (const float*)d_in[0];
  const float* x_item = (const float*)d_in[1];
  const int*   ei     = (const int*)d_in[2];
  const float* Ws     = (const float*)d_in[3];
  const float* Wt     = (const float*)d_in[4];
  const float* a      = (const float*)d_in[5];

  const int D      = 128;
  const int N_user = in_sizes[0] / D;
  const int N_item = in_sizes[1] / D;
  const int E      = in_sizes[2] / 2;
  const int* si = ei;
  const int* ti = ei + E;

  // Workspace layout: s | t | num | den
  float* s_   = (float*)d_ws;
  float* t_   = s_ + (size_t)N_user * D;
  float* num_ = t_ + (size_t)N_item * D;
  float* den_ = num_ + (size_t)N_item * D;

  // Zero accumulators every call (deterministic; capturable as memset node).
  hipMemsetAsync(num_, 0, ((size_t)N_item * D + N_item) * sizeof(float),
                 stream);

  // Projections: s = x_user @ Ws ; t = x_item @ Wt
  {
    dim3 blk(256);
    dim3 grd_s((N_user + 15) / 16);
    dim3 grd_t((N_item + 15) / 16);
    proj_gemm_kernel<<<grd_s, blk, 0, stream>>>(x_user, Ws, s_, N_user);
    proj_gemm_kernel<<<grd_t, blk, 0, stream>>>(x_item, Wt, t_, N_item);
  }

  // Per-edge attention + scatter (one wave per edge, 8 edges per block).
  {
    dim3 blk(256);
    dim3 grd((E + 7) / 8);
    edge_kernel<<<grd, blk, 0, stream>>>(s_, t_, si, ti, a, num_, den_, E);
  }

  // out = relu(t + num / (den + 1e-6))
  {
    const int total = N_item * D;
    dim3 blk(256);
    dim3 grd((total + 255) / 256);
    finalize_kernel<<<grd, blk, 0, stream>>>(t_, num_, den_, (float*)d_out,
                                             total);
  }
}